// StepwiseContextGuide_14628658610751
// MI455X (gfx1250) — compile-verified
//
#include <hip/hip_runtime.h>
#include <math.h>

typedef float v2f __attribute__((ext_vector_type(2)));
typedef float v8f __attribute__((ext_vector_type(8)));

#define HWN 4096      // H*W
#define NB  16        // batch
#define EPS 1e-5f

__device__ __forceinline__ float silu_f(float x) { return x / (1.f + expf(-x)); }

// ---------------------------------------------------------------------------
// Offset generation: grouped 3x3 conv (64 groups, 4 in-ch each) -> SiLU ->
// 1x1 conv (64 -> 2, with bias).  off: [B, 2, H, W]
// ---------------------------------------------------------------------------
__global__ __launch_bounds__(256)
void offset_kernel(const float* __restrict__ xl,   // [B,256,64,64]
                   const float* __restrict__ w1,   // [64,4,3,3]
                   const float* __restrict__ w2,   // [2,64]
                   const float* __restrict__ b2,   // [2]
                   float* __restrict__ off)        // [B,2,4096]
{
    __shared__ float w1s[2304];
    __shared__ float w2s[128];
    __shared__ float b2s[2];
    const int tid = threadIdx.x;
    for (int i = tid; i < 2304; i += 256) w1s[i] = w1[i];
    if (tid < 128) w2s[tid] = w2[tid];
    if (tid < 2)   b2s[tid] = b2[tid];
    __syncthreads();

    const int b = blockIdx.x >> 4;
    const int p = (blockIdx.x & 15) * 256 + tid;
    const int y = p >> 6, x = p & 63;

    float a0 = b2s[0], a1 = b2s[1];
    const float* xb = xl + (size_t)b * 256 * HWN;

    for (int g = 0; g < 64; ++g) {
        float s = 0.f;
        #pragma unroll
        for (int ic = 0; ic < 4; ++ic) {
            const float* ch  = xb  + (size_t)(g * 4 + ic) * HWN;
            const float* wch = w1s + (g * 4 + ic) * 9;
            #pragma unroll
            for (int dy = -1; dy <= 1; ++dy) {
                int yy = y + dy;
                if (yy < 0 || yy > 63) continue;
                #pragma unroll
                for (int dx = -1; dx <= 1; ++dx) {
                    int xx = x + dx;
                    if (xx < 0 || xx > 63) continue;
                    s += ch[yy * 64 + xx] * wch[(dy + 1) * 3 + (dx + 1)];
                }
            }
        }
        float ss = silu_f(s);
        a0 += w2s[g]      * ss;
        a1 += w2s[64 + g] * ss;
    }
    off[((size_t)b * 2 + 0) * HWN + p] = a0;
    off[((size_t)b * 2 + 1) * HWN + p] = a1;
}

// ---------------------------------------------------------------------------
// fp32 GEMM via V_WMMA_F32_16X16X4_F32.
//   C[m,n] = sum_k A[m,k] * B[b][k,n],  N = 4096 pixels per batch.
// Block: 256 threads = 8 waves; block tile 64(M) x 32(N); each wave one 16x16.
// MODE 0: dst = BN(C)                      (align branch -> g)
// MODE 1: dst = GELU(BN(C))                (mlp hidden   -> t)
// MODE 2: dst = SiLU(gridsample(g,off)+C)  (final output)
// ---------------------------------------------------------------------------
template <int MODE>
__global__ __launch_bounds__(256)
void gemm_wmma(const float* __restrict__ A,     // [M,K] row-major
               const float* __restrict__ Bm,    // [B,K,4096]
               float* __restrict__ dst,
               int M, int K,
               const float* __restrict__ bng, const float* __restrict__ bnb,
               const float* __restrict__ bnm, const float* __restrict__ bnv,
               const float* __restrict__ gbuf,  // MODE 2: [B,256,4096]
               const float* __restrict__ off)   // MODE 2: [B,2,4096]
{
    __shared__ float Alds[64][33];
    __shared__ float Blds[32][33];

    const int tid  = threadIdx.x;
    const int b    = blockIdx.z;
    const int mBlk = blockIdx.y * 64;
    const int nBlk = blockIdx.x * 32;
    const int wave = tid >> 5;
    const int lane = tid & 31;
    const int hi   = lane >> 4;          // 0: K+0,K+1   1: K+2,K+3
    const int lm   = lane & 15;
    const int wM   = (wave >> 1) * 16;   // 4 M-tiles
    const int wN   = (wave & 1) * 16;    // 2 N-tiles

    const float* Bb = Bm + (size_t)b * K * HWN;

    v8f acc = {0.f, 0.f, 0.f, 0.f, 0.f, 0.f, 0.f, 0.f};

    // A-load indices: 8 consecutive k per thread
    const int am = tid >> 2;             // 0..63
    const int ak = (tid & 3) * 8;        // 0,8,16,24
    // B-load indices: one float4 per thread
    const int bk = tid >> 3;             // 0..31
    const int bn = (tid & 7) * 4;        // 0..28

    for (int kc = 0; kc < K; kc += 32) {
        // stage A: 64x32
        {
            const float4* src = (const float4*)(A + (size_t)(mBlk + am) * K + kc + ak);
            float4 v0 = src[0];
            float4 v1 = src[1];
            float* d = &Alds[am][ak];
            d[0] = v0.x; d[1] = v0.y; d[2] = v0.z; d[3] = v0.w;
            d[4] = v1.x; d[5] = v1.y; d[6] = v1.z; d[7] = v1.w;
        }
        // stage B: 32x32
        {
            float4 v0 = *(const float4*)(Bb + (size_t)(kc + bk) * HWN + nBlk + bn);
            float* d = &Blds[bk][bn];
            d[0] = v0.x; d[1] = v0.y; d[2] = v0.z; d[3] = v0.w;
        }
        __syncthreads();

        #pragma unroll
        for (int kk = 0; kk < 32; kk += 4) {
            v2f av, bv;
            av.x = Alds[wM + lm][kk + 2 * hi + 0];
            av.y = Alds[wM + lm][kk + 2 * hi + 1];
            bv.x = Blds[kk + 2 * hi + 0][wN + lm];
            bv.y = Blds[kk + 2 * hi + 1][wN + lm];
            acc = __builtin_amdgcn_wmma_f32_16x16x4_f32(
                false, av, false, bv, (short)0, acc, false, false);
        }
        __syncthreads();
    }

    // Accumulator layout: VGPR i, lanes 0-15 -> (M=i,  N=lm); lanes 16-31 -> (M=i+8, N=lm)
    const int n = nBlk + wN + lm;        // pixel index 0..4095

    if constexpr (MODE == 0 || MODE == 1) {
        #pragma unroll
        for (int i = 0; i < 8; ++i) {
            int c = mBlk + wM + i + hi * 8;
            float sc = bng[c] * rsqrtf(bnv[c] + EPS);
            float sh = bnb[c] - bnm[c] * sc;
            float v  = acc[i] * sc + sh;
            if constexpr (MODE == 1)
                v = 0.5f * v * (1.f + erff(v * 0.70710678118654752f));
            dst[((size_t)b * M + c) * HWN + n] = v;
        }
    } else {
        // fused grid-sample(g) + add + SiLU.  All 8 outputs share pixel n.
        const int y = n >> 6, x = n & 63;
        float offx = off[((size_t)b * 2 + 0) * HWN + n];
        float offy = off[((size_t)b * 2 + 1) * HWN + n];
        // ix = ((gx+1)*W - 1)/2 with gx = -1 + 2x/63 + offx*2/64
        float ix = (float)x * (64.f / 63.f) - 0.5f + offx;
        float iy = (float)y * (64.f / 63.f) - 0.5f + offy;
        float x0f = floorf(ix), y0f = floorf(iy);
        float fx = ix - x0f, fy = iy - y0f;
        int x0 = (int)x0f, y0 = (int)y0f;
        int x1 = x0 + 1,   y1 = y0 + 1;
        bool vx0 = (x0 >= 0) && (x0 < 64), vx1 = (x1 >= 0) && (x1 < 64);
        bool vy0 = (y0 >= 0) && (y0 < 64), vy1 = (y1 >= 0) && (y1 < 64);
        int cx0 = min(max(x0, 0), 63), cx1 = min(max(x1, 0), 63);
        int cy0 = min(max(y0, 0), 63), cy1 = min(max(y1, 0), 63);
        float w00 = (1.f - fx) * (1.f - fy) * ((vx0 && vy0) ? 1.f : 0.f);
        float w10 = fx * (1.f - fy)         * ((vx1 && vy0) ? 1.f : 0.f);
        float w01 = (1.f - fx) * fy         * ((vx0 && vy1) ? 1.f : 0.f);
        float w11 = fx * fy                 * ((vx1 && vy1) ? 1.f : 0.f);
        int i00 = cy0 * 64 + cx0, i10 = cy0 * 64 + cx1;
        int i01 = cy1 * 64 + cx0, i11 = cy1 * 64 + cx1;

        #pragma unroll
        for (int i = 0; i < 8; ++i) {
            int c = mBlk + wM + i + hi * 8;
            const float* gp = gbuf + ((size_t)b * 256 + c) * HWN;
            float gs = w00 * gp[i00] + w10 * gp[i10]
                     + w01 * gp[i01] + w11 * gp[i11];
            float r = gs + acc[i];
            dst[((size_t)b * 256 + c) * HWN + n] = silu_f(r);
        }
    }
}

// ---------------------------------------------------------------------------
extern "C" void kernel_launch(void* const* d_in, const int* in_sizes, int n_in,
                              void* d_out, int out_size, void* d_ws, size_t ws_size,
                              hipStream_t stream) {
    const float* x_local = (const float*)d_in[0];
    const float* x_guide = (const float*)d_in[1];
    const float* w_align = (const float*)d_in[2];
    const float* bn1_g   = (const float*)d_in[3];
    const float* bn1_b   = (const float*)d_in[4];
    const float* bn1_m   = (const float*)d_in[5];
    const float* bn1_v   = (const float*)d_in[6];
    const float* w_off1  = (const float*)d_in[7];
    const float* w_off2  = (const float*)d_in[8];
    const float* b_off2  = (const float*)d_in[9];
    const float* w_mlp1  = (const float*)d_in[10];
    const float* bn2_g   = (const float*)d_in[11];
    const float* bn2_b   = (const float*)d_in[12];
    const float* bn2_m   = (const float*)d_in[13];
    const float* bn2_v   = (const float*)d_in[14];
    const float* w_mlp2  = (const float*)d_in[15];

    float* g   = (float*)d_ws;                              // [16,256,4096]
    float* t   = g + (size_t)NB * 256 * HWN;                // [16,128,4096]
    float* off = t + (size_t)NB * 128 * HWN;                // [16,2,4096]
    float* out = (float*)d_out;

    // 1) offsets (independent of GEMMs, launched first to overlap in queue)
    offset_kernel<<<NB * 16, 256, 0, stream>>>(x_local, w_off1, w_off2, b_off2, off);

    // 2) g = BN1(w_align @ x_guide)      M=256, K=256
    gemm_wmma<0><<<dim3(HWN / 32, 256 / 64, NB), 256, 0, stream>>>(
        w_align, x_guide, g, 256, 256, bn1_g, bn1_b, bn1_m, bn1_v, nullptr, nullptr);

    // 3) t = GELU(BN2(w_mlp1 @ x_local)) M=128, K=256
    gemm_wmma<1><<<dim3(HWN / 32, 128 / 64, NB), 256, 0, stream>>>(
        w_mlp1, x_local, t, 128, 256, bn2_g, bn2_b, bn2_m, bn2_v, nullptr, nullptr);

    // 4) out = SiLU(gridsample(g,off) + w_mlp2 @ t)   M=256, K=128
    gemm_wmma<2><<<dim3(HWN / 32, 256 / 64, NB), 256, 0, stream>>>(
        w_mlp2, t, out, 256, 128, nullptr, nullptr, nullptr, nullptr, g, off);
}